// PieceWiseActivation_69020124447017
// MI455X (gfx1250) — compile-verified
//
#include <hip/hip_runtime.h>

typedef float float4v __attribute__((ext_vector_type(4)));

// Piecewise-linear eval with tables resident in LDS.
// Matches: idx = clip(searchsorted(points, x, 'left') - 1, 0, n-2)
//          y = x*slope[idx] + intercept[idx]
//          out = x<=-10 ? 0 : (x>=10 ? x : y)
__device__ __forceinline__ float pw_eval(float xx, const float* __restrict__ s_pts,
                                         const float2* __restrict__ s_si, int npts) {
    int lo = 0, hi = npts;
    while (lo < hi) {
        int mid = (lo + hi) >> 1;
        if (s_pts[mid] < xx) lo = mid + 1; else hi = mid;
    }
    int idx = lo - 1;
    idx = idx < 0 ? 0 : idx;
    int mx = npts - 2;
    idx = idx > mx ? mx : idx;
    float2 si = s_si[idx];              // one ds_load_b64: (slope, intercept)
    float y = fmaf(xx, si.x, si.y);
    return (xx <= -10.0f) ? 0.0f : ((xx >= 10.0f) ? xx : y);
}

__global__ __launch_bounds__(256) void PieceWiseActivation_69020124447017_kernel(
    const float* __restrict__ x,
    const float* __restrict__ pts,
    const float* __restrict__ slopes,
    const float* __restrict__ icepts,
    float* __restrict__ out,
    long long n, int npts)
{
    extern __shared__ float smem[];
    float* s_pts = smem;                                  // npts floats
    // 8-byte aligned region for interleaved (slope, intercept) pairs
    float2* s_si = (float2*)(smem + ((npts + 1) & ~1));   // npts-1 float2

    for (int i = threadIdx.x; i < npts; i += blockDim.x)
        s_pts[i] = pts[i];
    for (int i = threadIdx.x; i < npts - 1; i += blockDim.x)
        s_si[i] = make_float2(slopes[i], icepts[i]);
    __syncthreads();   // gfx1250: s_barrier_signal -1 / s_barrier_wait -1

    const long long nvec  = n >> 2;
    long long tid    = (long long)blockIdx.x * blockDim.x + threadIdx.x;
    long long stride = (long long)gridDim.x * blockDim.x;

    const float4v* __restrict__ xv = (const float4v*)x;
    float4v* __restrict__ ov       = (float4v*)out;

    for (long long i = tid; i < nvec; i += stride) {
        // Prime next grid-stride chunk: global_prefetch_b8 (no LOADcnt cost)
        if (i + stride < nvec)
            __builtin_prefetch(&xv[i + stride], 0, 1);

        // Streaming 128-bit load: read-once data, 403MB working set > 192MB L2
        float4v v = __builtin_nontemporal_load(&xv[i]);
        float4v r;
#pragma unroll
        for (int k = 0; k < 4; ++k)
            r[k] = pw_eval(v[k], s_pts, s_si, npts);

        // Write-once output -> non-temporal 128-bit store
        __builtin_nontemporal_store(r, &ov[i]);
    }

    // Scalar tail (n % 4 != 0); no-op for the reference shape (n divisible by 4)
    const long long base = nvec << 2;
    for (long long i = base + tid; i < n; i += stride)
        out[i] = pw_eval(x[i], s_pts, s_si, npts);
}

extern "C" void kernel_launch(void* const* d_in, const int* in_sizes, int n_in,
                              void* d_out, int out_size, void* d_ws, size_t ws_size,
                              hipStream_t stream) {
    const float* x      = (const float*)d_in[0];
    const float* pts    = (const float*)d_in[1];
    const float* slopes = (const float*)d_in[2];
    const float* icepts = (const float*)d_in[3];
    float* out          = (float*)d_out;

    const long long n = (long long)in_sizes[0];
    const int npts    = in_sizes[1];

    const int block = 256;                         // 8 wave32 waves per block
    long long nvec  = n >> 2;
    long long want  = (nvec + block - 1) / block;
    int grid = (int)(want < 1 ? 1 : (want > 16384 ? 16384 : want));

    // LDS: points (padded to even count) + (npts-1) float2 pairs
    size_t shmem = (size_t)((npts + 1) & ~1) * sizeof(float)
                 + (size_t)(npts > 1 ? npts - 1 : 0) * sizeof(float2);

    PieceWiseActivation_69020124447017_kernel<<<grid, block, shmem, stream>>>(
        x, pts, slopes, icepts, out, n, npts);
}